// mtsnnMLP_63007170232440
// MI455X (gfx1250) — compile-verified
//
#include <hip/hip_runtime.h>

typedef float v2f __attribute__((ext_vector_type(2)));
typedef float v8f __attribute__((ext_vector_type(8)));

#define BETA  0.9f
#define THR   1.0f
#define NIN   2048
#define NHID  4096
#define NOUT  128
#define BTILE 32   // batch rows per workgroup (2 M-tiles of 16)

// D = A(16x4 f32) * B(4x16 f32) + C(16x16 f32), full fp32 precision
#define WMMA_F32(A, B, C) \
  __builtin_amdgcn_wmma_f32_16x16x4_f32(false, (A), false, (B), (short)0, (C), false, false)

__global__ __launch_bounds__(256, 1) void snn_fused_kernel(
    const float* __restrict__ x, const float* __restrict__ W1,
    const float* __restrict__ b1, const float* __restrict__ W2,
    const float* __restrict__ b2, float* __restrict__ out) {
  // 4-bit spike codes (one bit per timestep) for this block's 32-row strip
  __shared__ unsigned char spk[BTILE * NHID];  // 128 KB

  const int lane  = threadIdx.x & 31;
  const int wave  = threadIdx.x >> 5;        // 8 waves
  const int lm    = lane & 15;               // row/col within tile
  const int khalf = (lane >> 4) * 2;         // K sub-offset (0 or 2)
  const int hi8   = (lane >> 4) * 8;         // M offset for C/D halves
  const int row_base = blockIdx.x * BTILE;

  // ----------------- Phase 1: cur1 = x @ W1^T + b1 -> spike codes -----------------
  // 2 M-tiles x 4 N-tiles per wave-iteration: 6 loads feed 8 WMMAs
  const float* xrow0 = x + (size_t)(row_base +  0 + lm) * NIN + khalf;
  const float* xrow1 = x + (size_t)(row_base + 16 + lm) * NIN + khalf;

  for (int cg = 0; cg < 8; ++cg) {            // 8 x 64 = 512 hidden cols per wave
    const int col0 = (wave << 9) + (cg << 6);
    v8f a0 = {}, a1 = {}, a2 = {}, a3 = {};   // M-tile 0 accumulators
    v8f c0 = {}, c1 = {}, c2 = {}, c3 = {};   // M-tile 1 accumulators
    const float* w1p0 = W1 + (size_t)(col0 +  0 + lm) * NIN + khalf;
    const float* w1p1 = W1 + (size_t)(col0 + 16 + lm) * NIN + khalf;
    const float* w1p2 = W1 + (size_t)(col0 + 32 + lm) * NIN + khalf;
    const float* w1p3 = W1 + (size_t)(col0 + 48 + lm) * NIN + khalf;
#pragma unroll 2
    for (int k0 = 0; k0 < NIN; k0 += 4) {
      v2f av0 = *(const v2f*)(xrow0 + k0);    // A tiles: rows 0-15 / 16-31
      v2f av1 = *(const v2f*)(xrow1 + k0);
      v2f bb0 = *(const v2f*)(w1p0 + k0);     // B tiles: W1^T columns
      v2f bb1 = *(const v2f*)(w1p1 + k0);
      v2f bb2 = *(const v2f*)(w1p2 + k0);
      v2f bb3 = *(const v2f*)(w1p3 + k0);
      a0 = WMMA_F32(av0, bb0, a0);
      a1 = WMMA_F32(av0, bb1, a1);
      a2 = WMMA_F32(av0, bb2, a2);
      a3 = WMMA_F32(av0, bb3, a3);
      c0 = WMMA_F32(av1, bb0, c0);
      c1 = WMMA_F32(av1, bb1, c1);
      c2 = WMMA_F32(av1, bb2, c2);
      c3 = WMMA_F32(av1, bb3, c3);
    }

    // bias + 4-step LIF recurrence per element; reset_{t+1} == spk_t
#define SNN1_STORE(ACC, J, MT)                                            \
    {                                                                     \
      const int colj = col0 + (J) * 16;                                   \
      const float b1v = b1[colj + lm];                                    \
      _Pragma("unroll")                                                   \
      for (int r = 0; r < 8; ++r) {                                       \
        float c = (ACC)[r] + b1v;                                         \
        float m = c;                                                      \
        unsigned s = (m > THR) ? 1u : 0u;                                 \
        unsigned code = s;                                                \
        m = BETA * m + c - (s ? THR : 0.0f);                              \
        s = (m > THR) ? 1u : 0u;  code |= (s << 1);                       \
        m = BETA * m + c - (s ? THR : 0.0f);                              \
        s = (m > THR) ? 1u : 0u;  code |= (s << 2);                       \
        m = BETA * m + c - (s ? THR : 0.0f);                              \
        s = (m > THR) ? 1u : 0u;  code |= (s << 3);                       \
        spk[((MT)*16 + r + hi8) * NHID + colj + lm] = (unsigned char)code;\
      }                                                                   \
    }
    SNN1_STORE(a0, 0, 0)
    SNN1_STORE(a1, 1, 0)
    SNN1_STORE(a2, 2, 0)
    SNN1_STORE(a3, 3, 0)
    SNN1_STORE(c0, 0, 1)
    SNN1_STORE(c1, 1, 1)
    SNN1_STORE(c2, 2, 1)
    SNN1_STORE(c3, 3, 1)
#undef SNN1_STORE
  }

  __syncthreads();

  // --------- Phase 2: cur2_t = spk_t @ W2^T + b2 for t=1..4, mem2 recurrence ---------
  // Each wave: one 16-wide output slice, both M-tiles, all 4 timesteps:
  // per K=4 step: 2 ds_load_u16 + 1 global b64 feed 8 WMMAs.
  const int n0 = wave << 4;
  v8f d00 = {}, d01 = {}, d02 = {}, d03 = {};  // M-tile 0, t = 1..4
  v8f d10 = {}, d11 = {}, d12 = {}, d13 = {};  // M-tile 1, t = 1..4
  const float* w2p = W2 + (size_t)(n0 + lm) * NHID + khalf;
  const unsigned char* sp0 = spk + (      lm) * NHID + khalf;
  const unsigned char* sp1 = spk + (16 + lm) * NHID + khalf;
#pragma unroll 2
  for (int k0 = 0; k0 < NHID; k0 += 4) {
    unsigned cA = *(const unsigned short*)(sp0 + k0);  // 2 spike codes, M-tile 0
    unsigned cB = *(const unsigned short*)(sp1 + k0);  // 2 spike codes, M-tile 1
    v2f bw = *(const v2f*)(w2p + k0);
    unsigned cA0 = cA & 0xFFu, cA1 = cA >> 8;
    unsigned cB0 = cB & 0xFFu, cB1 = cB >> 8;
    v2f a;
    a.x = (float)( cA0       & 1u); a.y = (float)( cA1       & 1u); d00 = WMMA_F32(a, bw, d00);
    a.x = (float)((cA0 >> 1) & 1u); a.y = (float)((cA1 >> 1) & 1u); d01 = WMMA_F32(a, bw, d01);
    a.x = (float)((cA0 >> 2) & 1u); a.y = (float)((cA1 >> 2) & 1u); d02 = WMMA_F32(a, bw, d02);
    a.x = (float)((cA0 >> 3) & 1u); a.y = (float)((cA1 >> 3) & 1u); d03 = WMMA_F32(a, bw, d03);
    a.x = (float)( cB0       & 1u); a.y = (float)( cB1       & 1u); d10 = WMMA_F32(a, bw, d10);
    a.x = (float)((cB0 >> 1) & 1u); a.y = (float)((cB1 >> 1) & 1u); d11 = WMMA_F32(a, bw, d11);
    a.x = (float)((cB0 >> 2) & 1u); a.y = (float)((cB1 >> 2) & 1u); d12 = WMMA_F32(a, bw, d12);
    a.x = (float)((cB0 >> 3) & 1u); a.y = (float)((cB1 >> 3) & 1u); d13 = WMMA_F32(a, bw, d13);
  }

  const float b2v = b2[n0 + lm];
#define SNN2_STORE(D1, D2, D3, D4, MT)                                    \
  {                                                                       \
    float* orow = out + (size_t)(row_base + (MT)*16 + hi8) * NOUT + n0 + lm; \
    _Pragma("unroll")                                                     \
    for (int r = 0; r < 8; ++r) {                                         \
      float c1_ = (D1)[r] + b2v;                                          \
      float c2_ = (D2)[r] + b2v;                                          \
      float c3_ = (D3)[r] + b2v;                                          \
      float c4_ = (D4)[r] + b2v;                                          \
      float m = c1_;                      /* mem2_1 (mem2_0 = 0) */       \
      float sum = m;                                                      \
      float rs = (m > THR) ? THR : 0.0f;                                  \
      m = BETA * m + c2_ - rs; sum += m;                                  \
      rs = (m > THR) ? THR : 0.0f;                                        \
      m = BETA * m + c3_ - rs; sum += m;                                  \
      rs = (m > THR) ? THR : 0.0f;                                        \
      m = BETA * m + c4_ - rs; sum += m;                                  \
      orow[(size_t)r * NOUT] = sum * 0.25f;  /* mean over timesteps */    \
    }                                                                     \
  }
  SNN2_STORE(d00, d01, d02, d03, 0)
  SNN2_STORE(d10, d11, d12, d13, 1)
#undef SNN2_STORE
}

extern "C" void kernel_launch(void* const* d_in, const int* in_sizes, int n_in,
                              void* d_out, int out_size, void* d_ws, size_t ws_size,
                              hipStream_t stream) {
  (void)n_in; (void)out_size; (void)d_ws; (void)ws_size;
  const float* x  = (const float*)d_in[0];
  const float* W1 = (const float*)d_in[1];
  const float* b1 = (const float*)d_in[2];
  const float* W2 = (const float*)d_in[3];
  const float* b2 = (const float*)d_in[4];
  float* out = (float*)d_out;

  const int B = in_sizes[0] / NIN;     // 8192
  dim3 grid(B / BTILE);                // 256 workgroups of 32 batch rows
  dim3 block(256);                     // 8 wave32 per workgroup
  snn_fused_kernel<<<grid, block, 0, stream>>>(x, W1, b1, W2, b2, out);
}